// ColorHistogram_17248588661414
// MI455X (gfx1250) — compile-verified
//
#include <hip/hip_runtime.h>

// ---------------------------------------------------------------------------
// Soft color histogram (L,a,b channels, 64 Gaussian bins each) for gfx1250.
//
// Core trick: exponent(-k*(p-c)^2) is a rank-3 bilinear form
//   [-k*p^2] + [2k*c]*p + [-k*c^2]  =  [p^2, p, 1, 0] . [-k, 2kc, -kc^2, 0]
// so a V_WMMA_F32_16X16X4_F32 with A = 16 pixels x [p^2,p,1,0] and
// B = coeffs x 16 bins produces 256 base-2 exponents per instruction,
// moving 3 of the 5 per-weight VALU ops onto the matrix pipe.
// ---------------------------------------------------------------------------

typedef float v2f __attribute__((ext_vector_type(2)));
typedef float v8f __attribute__((ext_vector_type(8)));

#define SPX (512 * 512)          // pixels per channel image
#define NBATCH 16
#define NBINS 64
#define WAVES_PER_IMG 256
#define PPW (SPX / WAVES_PER_IMG)     // 1024 pixels per wave
#define PIX_PER_ITER 128              // 32 lanes * float4
#define ITERS (PPW / PIX_PER_ITER)    // 8
#define WAVES_PER_BLOCK 4
#define BLOCK (WAVES_PER_BLOCK * 32)
#define NBLOCKS (NBATCH * WAVES_PER_IMG / WAVES_PER_BLOCK)  // 1024

__device__ __forceinline__ float fexp2(float x) { return __builtin_amdgcn_exp2f(x); }
__device__ __forceinline__ float flog2(float x) { return __builtin_amdgcn_logf(x); }  // v_log_f32 = log2

__device__ __forceinline__ float srgb_lin(float x) {
  // pow(u, 2.4) via exp2(2.4*log2(u)); branch arm with log2(<=0) is discarded by select
  float p = fexp2(2.4f * flog2((x + 0.055f) * (1.0f / 1.055f)));
  return (x > 0.04045f) ? p : x * (1.0f / 12.92f);
}
__device__ __forceinline__ float lab_f(float t) {
  float c = fexp2(flog2(t) * (1.0f / 3.0f));  // cbrt for t > eps
  return (t > 0.008856f) ? c : (7.787f * t + 16.0f / 116.0f);
}

__global__ __launch_bounds__(BLOCK) void hist_accum_kernel(const float* __restrict__ img,
                                                           float* __restrict__ out) {
  const int lane   = threadIdx.x & 31;
  const int wib    = threadIdx.x >> 5;
  const int gwave  = blockIdx.x * WAVES_PER_BLOCK + wib;
  const int batch  = gwave >> 8;           // / WAVES_PER_IMG
  const int wimg   = gwave & (WAVES_PER_IMG - 1);
  const float* base = img + (size_t)batch * 3 * SPX;
  const int p0 = wimg * PPW;

  const float log2e = 1.44269504088896340736f;
  // channel params: L: [0,100], a/b: [-128,127], sigma = 0.5*delta
  const float deltaA[3] = {100.0f / 64.0f, 255.0f / 64.0f, 255.0f / 64.0f};
  const float vminA[3]  = {0.0f, -128.0f, -128.0f};
  const float midA[3]   = {50.0f, -0.5f, -0.5f};  // center values to reduce cancellation error
  float K2[3];
#pragma unroll
  for (int ch = 0; ch < 3; ++ch) {
    float sig = 0.5f * deltaA[ch];
    K2[ch] = log2e / (2.0f * sig * sig);   // exponent already in log2 units -> use exp2
  }

  // Constant B operands: 4 (K) x 16 (bins) per bin-group.
  // f32 B 4x16 layout (mirrors documented A 16x4): lanes 0-15 hold K=0 (VGPR0)
  // and K=1 (VGPR1) for column N=lane; lanes 16-31 hold K=2,3 for N=lane-16.
  v2f Bc[3][4];
#pragma unroll
  for (int ch = 0; ch < 3; ++ch) {
#pragma unroll
    for (int g = 0; g < 4; ++g) {
      int n = g * 16 + (lane & 15);
      float c = vminA[ch] + deltaA[ch] * ((float)n + 0.5f) - midA[ch];
      float b0 = (lane < 16) ? (-K2[ch]) : (-K2[ch] * c * c);  // K=0 row : K=2 row
      float b1 = (lane < 16) ? (2.0f * K2[ch] * c) : 0.0f;     // K=1 row : K=3 row
      v2f b; b.x = b0; b.y = b1;
      Bc[ch][g] = b;
    }
  }

  // Accumulator tiles: 16 pixels x 16 bins each, 3 channels x 4 bin groups.
  v8f acc[3][4];
  const v8f vzero = {0.f, 0.f, 0.f, 0.f, 0.f, 0.f, 0.f, 0.f};
#pragma unroll
  for (int ch = 0; ch < 3; ++ch)
#pragma unroll
    for (int g = 0; g < 4; ++g) acc[ch][g] = vzero;

  for (int it = 0; it < ITERS; ++it) {
    const int p = p0 + it * PIX_PER_ITER + lane * 4;
    const float4 r4 = *reinterpret_cast<const float4*>(base + p);
    const float4 g4 = *reinterpret_cast<const float4*>(base + p + SPX);
    const float4 b4 = *reinterpret_cast<const float4*>(base + p + 2 * SPX);
    if (it + 1 < ITERS) {  // gfx1250 global_prefetch_b8 path
      __builtin_prefetch(base + p + PIX_PER_ITER, 0, 0);
      __builtin_prefetch(base + p + SPX + PIX_PER_ITER, 0, 0);
      __builtin_prefetch(base + p + 2 * SPX + PIX_PER_ITER, 0, 0);
    }
    const float rr[4] = {r4.x, r4.y, r4.z, r4.w};
    const float gg[4] = {g4.x, g4.y, g4.z, g4.w};
    const float bb[4] = {b4.x, b4.y, b4.z, b4.w};

#pragma unroll
    for (int j = 0; j < 4; ++j) {
      float r = srgb_lin(rr[j]);
      float g = srgb_lin(gg[j]);
      float b = srgb_lin(bb[j]);
      float x = 0.412453f * r + 0.35758f * g + 0.180423f * b;
      float y = 0.212671f * r + 0.71516f * g + 0.072169f * b;
      float z = 0.019334f * r + 0.119193f * g + 0.950227f * b;
      x *= (1.0f / 0.950456f);
      z *= (1.0f / 1.088754f);
      float fx = lab_f(x), fy = lab_f(y), fz = lab_f(z);
      float chv[3];
      chv[0] = 116.0f * fy - 16.0f - midA[0];
      chv[1] = 500.0f * (fx - fy) - midA[1];
      chv[2] = 200.0f * (fy - fz) - midA[2];

#pragma unroll
      for (int ch = 0; ch < 3; ++ch) {
        float v0 = chv[ch];
        float v1 = __shfl_xor(v0, 16, 32);  // other half's pixel value
#pragma unroll
        for (int h = 0; h < 2; ++h) {
          float v = h ? v1 : v0;
          // A 16x4: lanes 0-15 VGPR0=K0 (p^2), VGPR1=K1 (p);
          //         lanes 16-31 VGPR0=K2 (1), VGPR1=K3 (0)   [documented layout]
          v2f A;
          A.x = (lane < 16) ? (v * v) : 1.0f;
          A.y = (lane < 16) ? v : 0.0f;
#pragma unroll
          for (int g2 = 0; g2 < 4; ++g2) {
            v8f e = __builtin_amdgcn_wmma_f32_16x16x4_f32(
                false, A, false, Bc[ch][g2], (short)0, vzero, false, false);
            v8f w;
#pragma unroll
            for (int k = 0; k < 8; ++k) w[k] = fexp2(e[k]);
            acc[ch][g2] += w;
          }
        }
      }
    }
  }

  // Reduce pixel (M) dimension: 8 VGPRs (M=0..7 / 8..15 per lane half) + xor-16.
#pragma unroll
  for (int ch = 0; ch < 3; ++ch) {
#pragma unroll
    for (int g2 = 0; g2 < 4; ++g2) {
      v8f t = acc[ch][g2];
      float s = ((t[0] + t[1]) + (t[2] + t[3])) + ((t[4] + t[5]) + (t[6] + t[7]));
      s += __shfl_xor(s, 16, 32);
      if (lane < 16) {
        atomicAdd(&out[batch * 192 + ch * 64 + g2 * 16 + lane], s);
      }
    }
  }
}

__global__ void zero_out_kernel(float* __restrict__ out, int n) {
  int i = blockIdx.x * blockDim.x + threadIdx.x;
  if (i < n) out[i] = 0.0f;
}

// Normalize each 64-bin channel histogram in place: h / (sum(h) + 1e-8)
__global__ void finalize_kernel(float* __restrict__ out) {
  const int b = blockIdx.x;        // batch
  const int t = threadIdx.x;       // 0..191
  const int ch = t >> 6;
  const float* h = out + b * 192 + ch * 64;
  float sum = 0.0f;
#pragma unroll
  for (int n = 0; n < 64; ++n) sum += h[n];
  float val = out[b * 192 + t];
  __syncthreads();  // all reads complete before in-place writes
  out[b * 192 + t] = val / (sum + 1e-8f);
}

extern "C" void kernel_launch(void* const* d_in, const int* in_sizes, int n_in,
                              void* d_out, int out_size, void* d_ws, size_t ws_size,
                              hipStream_t stream) {
  (void)in_sizes; (void)n_in; (void)d_ws; (void)ws_size;
  const float* img = (const float*)d_in[0];
  float* out = (float*)d_out;

  zero_out_kernel<<<(out_size + 255) / 256, 256, 0, stream>>>(out, out_size);
  hist_accum_kernel<<<NBLOCKS, BLOCK, 0, stream>>>(img, out);
  finalize_kernel<<<NBATCH, 192, 0, stream>>>(out);
}